// GAT_8787503088331
// MI455X (gfx1250) — compile-verified
//
#include <hip/hip_runtime.h>
#include <hip/hip_bf16.h>

typedef __attribute__((ext_vector_type(2))) float v2f;
typedef __attribute__((ext_vector_type(8))) float v8f;

#define HC 256   // H*C
#define INCH 128 // in channels
#define NEG_SLOPE 0.2f

// ---------------------------------------------------------------------------
// init: zero the feature accumulator, set m=-inf, z=0
// ---------------------------------------------------------------------------
__global__ __launch_bounds__(256) void k_init(float* __restrict__ acc,
                                              float* __restrict__ m_,
                                              float* __restrict__ zz, int N) {
    int t = blockIdx.x * 256 + threadIdx.x;
    if (t < N * HC) acc[t] = 0.0f;
    if (t < N * 4) {
        m_[t] = __int_as_float(0xff800000); // -inf
        zz[t] = 0.0f;
    }
}

// ---------------------------------------------------------------------------
// GEMM1: h[N,256] = x[N,128] @ W[128,256]   (fp32 WMMA 16x16x4)
// grid: (ceil(N/64), 16), block: 128 (4 waves, each wave owns a 16x16 tile)
// ---------------------------------------------------------------------------
__global__ __launch_bounds__(128) void k_gemm_xw(const float* __restrict__ x,
                                                 const float* __restrict__ W,
                                                 float* __restrict__ h, int N) {
    __shared__ float Bs[INCH * 16]; // [k][n] tile of W, 8KB
    const int tid = threadIdx.x;
    const int n0  = blockIdx.y * 16;
    for (int idx = tid; idx < INCH * 16; idx += 128) {
        int k = idx >> 4, n = idx & 15;
        Bs[idx] = W[k * HC + n0 + n];
    }
    __syncthreads();

    const int lane = tid & 31;
    const int wave = tid >> 5;
    const int rows = blockIdx.x * 64 + wave * 16;
    int row = rows + (lane & 15);
    if (row >= N) row = N - 1;
    const int koff = (lane >> 4) * 2;      // lanes 0-15: K={0,1}; 16-31: K={2,3}
    const int nsel = lane & 15;
    const float* xr = x + (size_t)row * INCH;

    v8f c = {};
    for (int kk = 0; kk < INCH / 4; ++kk) {
        int k = kk * 4 + koff;
        v2f a; a.x = xr[k]; a.y = xr[k + 1];
        v2f b; b.x = Bs[k * 16 + nsel]; b.y = Bs[(k + 1) * 16 + nsel];
        c = __builtin_amdgcn_wmma_f32_16x16x4_f32(false, a, false, b,
                                                  (short)0, c, false, false);
    }
    const int col   = n0 + nsel;
    const int rbase = rows + ((lane >> 4) << 3); // D: vgpr r -> M=r (+8 for hi half)
    for (int r = 0; r < 8; ++r) {
        int ro = rbase + r;
        if (ro < N) h[(size_t)ro * HC + col] = c[r];
    }
}

// ---------------------------------------------------------------------------
// attention dots: a_s[n,h]=sum_c h[n,h,c]*att_src[h,c] (same for a_d)
// one wave per node; lane l covers channels [8l,8l+8) -> head = l>>3
// ---------------------------------------------------------------------------
__global__ __launch_bounds__(256) void k_att(const float* __restrict__ h,
                                             const float* __restrict__ att_s,
                                             const float* __restrict__ att_d,
                                             float* __restrict__ a_s,
                                             float* __restrict__ a_d, int N) {
    int gw   = (blockIdx.x * 256 + threadIdx.x) >> 5;
    int lane = threadIdx.x & 31;
    if (gw >= N) return;
    const float4* hr  = (const float4*)(h + (size_t)gw * HC);
    const float4* as4 = (const float4*)att_s;
    const float4* ad4 = (const float4*)att_d;
    float ss = 0.f, sd = 0.f;
    for (int j = 0; j < 2; ++j) {
        int i = lane * 2 + j;
        float4 v = hr[i], s = as4[i], d = ad4[i];
        ss += v.x * s.x + v.y * s.y + v.z * s.z + v.w * s.w;
        sd += v.x * d.x + v.y * d.y + v.z * d.z + v.w * d.w;
    }
    for (int off = 1; off < 8; off <<= 1) { // reduce within 8-lane head groups
        ss += __shfl_xor(ss, off, 32);
        sd += __shfl_xor(sd, off, 32);
    }
    if ((lane & 7) == 0) {
        int head = lane >> 3;
        a_s[gw * 4 + head] = ss;
        a_d[gw * 4 + head] = sd;
    }
}

// ---------------------------------------------------------------------------
// float atomic max via signed-max / unsigned-min trick
// ---------------------------------------------------------------------------
__device__ inline void atomicMaxFloat(float* addr, float val) {
    if (!(val < 0.0f))
        atomicMax((int*)addr, __float_as_int(val));
    else
        atomicMin((unsigned int*)addr, (unsigned int)__float_as_int(val));
}

__device__ inline void edge_src_dst(const int* __restrict__ ei, int e, int E,
                                    int& src, int& dst) {
    if (e < E) { src = ei[e]; dst = ei[E + e]; }
    else       { src = e - E; dst = e - E; }     // self-loops appended
}

// ---------------------------------------------------------------------------
// pass A: segment max over destinations
// ---------------------------------------------------------------------------
__global__ __launch_bounds__(256) void k_edge_max(const int* __restrict__ ei,
                                                  const float* __restrict__ a_s,
                                                  const float* __restrict__ a_d,
                                                  float* __restrict__ m_,
                                                  int E, int E2) {
    int e = blockIdx.x * 256 + threadIdx.x;
    if (e >= E2) return;
    int src, dst;
    edge_src_dst(ei, e, E, src, dst);
    for (int hh = 0; hh < 4; ++hh) {
        float v = a_s[src * 4 + hh] + a_d[dst * 4 + hh];
        v = v > 0.f ? v : NEG_SLOPE * v;
        atomicMaxFloat(&m_[dst * 4 + hh], v);
    }
}

// ---------------------------------------------------------------------------
// pass B: ex = exp(e - m[dst]); z[dst] += ex
// ---------------------------------------------------------------------------
__global__ __launch_bounds__(256) void k_edge_sum(const int* __restrict__ ei,
                                                  const float* __restrict__ a_s,
                                                  const float* __restrict__ a_d,
                                                  const float* __restrict__ m_,
                                                  float* __restrict__ ex,
                                                  float* __restrict__ zz,
                                                  int E, int E2) {
    int e = blockIdx.x * 256 + threadIdx.x;
    if (e >= E2) return;
    int src, dst;
    edge_src_dst(ei, e, E, src, dst);
    for (int hh = 0; hh < 4; ++hh) {
        float v = a_s[src * 4 + hh] + a_d[dst * 4 + hh];
        v = v > 0.f ? v : NEG_SLOPE * v;
        float x = expf(v - m_[dst * 4 + hh]);
        ex[(size_t)e * 4 + hh] = x;
        atomicAdd(&zz[dst * 4 + hh], x);
    }
}

// ---------------------------------------------------------------------------
// pass C: acc[dst,:] += alpha * h[src,:]  (float4 per thread; atomics hit L2)
// ---------------------------------------------------------------------------
__global__ __launch_bounds__(256) void k_scatter(const int* __restrict__ ei,
                                                 const float* __restrict__ h,
                                                 const float* __restrict__ ex,
                                                 const float* __restrict__ zz,
                                                 float* __restrict__ acc,
                                                 int E, int E2) {
    long long t     = (long long)blockIdx.x * 256 + threadIdx.x;
    long long total = (long long)E2 * 64;
    if (t >= total) return;
    int e = (int)(t >> 6);
    int g = (int)(t & 63);            // float4 group within 256-ch row
    int src, dst;
    edge_src_dst(ei, e, E, src, dst);
    int head = g >> 4;
    float alpha = ex[(size_t)e * 4 + head] / (zz[dst * 4 + head] + 1e-16f);
    float4 hv = *(const float4*)(h + (size_t)src * HC + g * 4);
    float* o  = acc + (size_t)dst * HC + g * 4;
    atomicAdd(o + 0, alpha * hv.x);
    atomicAdd(o + 1, alpha * hv.y);
    atomicAdd(o + 2, alpha * hv.z);
    atomicAdd(o + 3, alpha * hv.w);
}

// ---------------------------------------------------------------------------
// GEMM2: y[N,64] = elu(acc + bias) @ fc_w^T + fc_b   (fp32 WMMA, K=256)
// block 256 = 8 waves: 2 M-tiles x 4 N-tiles = 32 rows x 64 cols per WG
// fc_w staged in LDS (64KB) with add-swizzle: conflict-free load and read
// ---------------------------------------------------------------------------
__global__ __launch_bounds__(256) void k_out(const float* __restrict__ acc,
                                             const float* __restrict__ bias,
                                             const float* __restrict__ fc_w,
                                             const float* __restrict__ fc_b,
                                             float* __restrict__ y, int N) {
    __shared__ float Bs[HC * 64]; // B[k][n] = fc_w[n*256+k], swizzled
    const int tid = threadIdx.x;
    for (int idx = tid; idx < 64 * HC; idx += 256) {
        int n = idx >> 8, k = idx & 255;
        Bs[k * 64 + ((n + k) & 63)] = fc_w[idx];
    }
    __syncthreads();

    const int lane = tid & 31;
    const int wave = tid >> 5;
    const int rows = blockIdx.x * 32 + (wave >> 2) * 16;
    const int n0   = (wave & 3) * 16;
    int row = rows + (lane & 15);
    if (row >= N) row = N - 1;
    const int koff = (lane >> 4) * 2;
    const int nn   = n0 + (lane & 15);
    const float* ar = acc + (size_t)row * HC;

    v8f c = {};
    for (int kk = 0; kk < HC / 4; ++kk) {
        int k = kk * 4 + koff;
        float a0 = ar[k] + bias[k];
        float a1 = ar[k + 1] + bias[k + 1];
        a0 = a0 > 0.f ? a0 : (expf(a0) - 1.f); // ELU
        a1 = a1 > 0.f ? a1 : (expf(a1) - 1.f);
        v2f a; a.x = a0; a.y = a1;
        v2f b;
        b.x = Bs[k * 64 + ((nn + k) & 63)];
        b.y = Bs[(k + 1) * 64 + ((nn + k + 1) & 63)];
        c = __builtin_amdgcn_wmma_f32_16x16x4_f32(false, a, false, b,
                                                  (short)0, c, false, false);
    }
    const int rbase = rows + ((lane >> 4) << 3);
    const float fb  = fc_b[nn];
    for (int r = 0; r < 8; ++r) {
        int ro = rbase + r;
        if (ro < N) y[(size_t)ro * 64 + nn] = c[r] + fb;
    }
}

// ---------------------------------------------------------------------------
extern "C" void kernel_launch(void* const* d_in, const int* in_sizes, int n_in,
                              void* d_out, int out_size, void* d_ws, size_t ws_size,
                              hipStream_t stream) {
    const float* x     = (const float*)d_in[0];
    const float* W     = (const float*)d_in[1];
    const float* att_s = (const float*)d_in[2];
    const float* att_d = (const float*)d_in[3];
    const float* bias  = (const float*)d_in[4];
    const float* fc_w  = (const float*)d_in[5];
    const float* fc_b  = (const float*)d_in[6];
    const int*   ei    = (const int*)d_in[7];

    const int N  = in_sizes[0] / INCH;
    const int E  = in_sizes[7] / 2;
    const int E2 = E + N;
    float* y = (float*)d_out;

    // workspace carve-up (~119 MB total)
    float* h   = (float*)d_ws;                 // N*256
    float* acc = h   + (size_t)N * HC;         // N*256
    float* a_s = acc + (size_t)N * HC;         // N*4
    float* a_d = a_s + (size_t)N * 4;          // N*4
    float* m_  = a_d + (size_t)N * 4;          // N*4
    float* zz  = m_  + (size_t)N * 4;          // N*4
    float* ex  = zz  + (size_t)N * 4;          // E2*4

    k_init<<<(N * HC + 255) / 256, 256, 0, stream>>>(acc, m_, zz, N);

    dim3 g1((N + 63) / 64, HC / 16);
    k_gemm_xw<<<g1, 128, 0, stream>>>(x, W, h, N);

    k_att<<<(N + 7) / 8, 256, 0, stream>>>(h, att_s, att_d, a_s, a_d, N);

    int eb = (E2 + 255) / 256;
    k_edge_max<<<eb, 256, 0, stream>>>(ei, a_s, a_d, m_, E, E2);
    k_edge_sum<<<eb, 256, 0, stream>>>(ei, a_s, a_d, m_, ex, zz, E, E2);

    long long total = (long long)E2 * 64;
    int sb = (int)((total + 255) / 256);
    k_scatter<<<sb, 256, 0, stream>>>(ei, h, ex, zz, acc, E, E2);

    k_out<<<(N + 31) / 32, 256, 0, stream>>>(acc, bias, fc_w, fc_b, y, N);
}